// Decomposed_LC_41858751267220
// MI455X (gfx1250) — compile-verified
//
#include <hip/hip_runtime.h>
#include <stdint.h>

typedef __attribute__((ext_vector_type(2))) float v2f;
typedef __attribute__((ext_vector_type(8))) float v8f;

#define B_SZ  128
#define CIN   32
#define COUT  64
#define OHW   32
#define HW    34   // OH + KH - 1
#define KTAPS 9

// Per-lane async global->LDS dword copy (GVS mode: SGPR64 base + VGPR32 byte
// offset; vdst VGPR holds the LDS byte address). Tracked by ASYNCcnt.
__device__ __forceinline__ void async_ld_dword(uint32_t lds_addr,
                                               uint32_t goff_bytes,
                                               const float* sbase) {
  asm volatile("global_load_async_to_lds_b32 %0, %1, %2"
               :: "v"(lds_addr), "v"(goff_bytes), "s"(sbase)
               : "memory");
}
__device__ __forceinline__ void wait_async0() {
  asm volatile("s_wait_asynccnt 0" ::: "memory");
}

// One workgroup per output position (i,j): GEMM M=128 (batch), N=64 (cout),
// K=288 (cin*9), processed tap-major in 9 chunks of 32.
// 8 waves; wave w -> M-tile w, each wave computes all 4 N-tiles.
__global__ __launch_bounds__(256) void lc_wmma_f32_kernel(
    const float* __restrict__ x,      // (128, 32, 34, 34)
    const float* __restrict__ wgt,    // (64, 32, 32, 32, 9)
    const float* __restrict__ bias,   // (64, 32, 32)
    float* __restrict__ out)          // (128, 64, 32, 32)
{
  // +2 pad -> row stride 34 dwords: conflict-free ds_load_b64 fragment reads
  __shared__ float As[2][B_SZ][CIN + 2];   // 2*128*34*4 = 34816 B
  __shared__ float Bs[2][COUT][CIN + 2];   // 2* 64*34*4 = 17408 B

  const int j      = blockIdx.x;
  const int i      = blockIdx.y;
  const int tid    = threadIdx.x;
  const int wave   = tid >> 5;
  const int lane   = tid & 31;
  const int lane16 = lane & 15;
  const int lhalf  = lane >> 4;
  const int m_base = wave * 16;

  const int posoff9 = (i * OHW + j) * 9;   // per-position weight offset (elems)

  // Issue all async global->LDS copies for one tap into LDS buffer `buf`.
  auto issue_tap = [&](int buf, int tap) {
    const int kh = tap / 3, kw = tap - 3 * (tap / 3);
    const uint32_t xoff = (uint32_t)(((i + kh) * HW + (j + kw)) * 4);
    #pragma unroll
    for (int q = 0; q < 16; ++q) {
      const int e = tid + q * 256;                    // e = b*32 + c
      const uint32_t lds = (uint32_t)(uintptr_t)&As[buf][e >> 5][e & 31];
      async_ld_dword(lds, (uint32_t)e * (HW * HW * 4) + xoff, x);
    }
    const uint32_t woff = (uint32_t)((posoff9 + tap) * 4);
    #pragma unroll
    for (int q = 0; q < 8; ++q) {
      const int e = tid + q * 256;                    // e = o*32 + c
      const uint32_t lds = (uint32_t)(uintptr_t)&Bs[buf][e >> 5][e & 31];
      async_ld_dword(lds, (uint32_t)e * (OHW * OHW * 9 * 4) + woff, wgt);
    }
  };

  v8f acc[4] = {};

  issue_tap(0, 0);
  wait_async0();
  __syncthreads();

  for (int tap = 0; tap < KTAPS; ++tap) {
    const int cur = tap & 1;
    if (tap + 1 < KTAPS) issue_tap(cur ^ 1, tap + 1);  // overlap with WMMA

    #pragma unroll
    for (int t = 0; t < 8; ++t) {                // 8 k-steps of K=4
      const int kc = t * 4 + lhalf * 2;          // A lanes: K = 2*(lane/16)+{0,1}
      const v2f a = *(const v2f*)&As[cur][m_base + lane16][kc];
      #pragma unroll
      for (int n = 0; n < 4; ++n) {
        const v2f b = *(const v2f*)&Bs[cur][n * 16 + lane16][kc];
        acc[n] = __builtin_amdgcn_wmma_f32_16x16x4_f32(
            false, a, false, b, (short)0, acc[n], false, false);
      }
    }

    if (tap + 1 < KTAPS) wait_async0();          // next buffer fully in LDS
    __syncthreads();
  }

  // Epilogue: D layout — lane L holds N = L%16, M = vgpr + 8*(L/16)
  #pragma unroll
  for (int n = 0; n < 4; ++n) {
    const int o  = n * 16 + lane16;
    const float bv = bias[(o * OHW + i) * OHW + j];
    #pragma unroll
    for (int vr = 0; vr < 8; ++vr) {
      const int m = m_base + lhalf * 8 + vr;
      out[((m * COUT + o) * OHW + i) * OHW + j] = acc[n][vr] + bv;
    }
  }
}

extern "C" void kernel_launch(void* const* d_in, const int* in_sizes, int n_in,
                              void* d_out, int out_size, void* d_ws, size_t ws_size,
                              hipStream_t stream) {
  const float* x    = (const float*)d_in[0];
  const float* wgt  = (const float*)d_in[1];
  const float* bias = (const float*)d_in[2];
  float* out        = (float*)d_out;

  dim3 grid(OHW, OHW);   // one workgroup per (i,j)
  lc_wmma_f32_kernel<<<grid, 256, 0, stream>>>(x, wgt, bias, out);
}